// GCN_78176994721837
// MI455X (gfx1250) — compile-verified
//
#include <hip/hip_runtime.h>
#include <hip/hip_bf16.h>

typedef __attribute__((ext_vector_type(2))) float v2f;
typedef __attribute__((ext_vector_type(8))) float v8f;
typedef __attribute__((ext_vector_type(4))) unsigned int v4u;
typedef __attribute__((ext_vector_type(8))) int v8i;
typedef __attribute__((ext_vector_type(4))) int v4i;

// ---------------- degree / normalization ----------------

__global__ void gcn_init_deg(float* deg, int n) {
    int i = blockIdx.x * blockDim.x + threadIdx.x;
    if (i < n) deg[i] = 1.0f;   // self-loop
}

__global__ void gcn_acc_deg(const int* __restrict__ dst, float* deg, int e) {
    int i = blockIdx.x * blockDim.x + threadIdx.x;
    if (i < e) atomicAdd(&deg[dst[i]], 1.0f);
}

__global__ void gcn_make_dinv(float* deg, int n) {
    int i = blockIdx.x * blockDim.x + threadIdx.x;
    if (i < n) deg[i] = rsqrtf(deg[i]);   // deg >= 1 always (self-loop)
}

// ---------------- pad W3 from [K,C] to [K,Cpad] ----------------

__global__ void gcn_pad_w(const float* __restrict__ w, float* __restrict__ wp,
                          int k, int c, int cpad) {
    int i = blockIdx.x * blockDim.x + threadIdx.x;
    if (i < k * cpad) {
        int r = i / cpad, col = i - r * cpad;
        wp[i] = (col < c) ? w[r * c + col] : 0.0f;
    }
}

// ---------------- TDM: linear global -> LDS copy of nelem f32 ----------------

#if __has_builtin(__builtin_amdgcn_tensor_load_to_lds)
__device__ __forceinline__ void tdm_load_linear_to_lds(const void* gptr,
                                                       unsigned lds_off,
                                                       unsigned nelem) {
    unsigned long long ga = (unsigned long long)(uintptr_t)gptr;
    v4u g0;
    g0[0] = 1u;                                   // count=1 (valid), user mode
    g0[1] = lds_off;                              // lds_addr (bytes)
    g0[2] = (unsigned)(ga & 0xFFFFFFFFu);         // global_addr[31:0]
    g0[3] = (unsigned)((ga >> 32) & 0x01FFFFFFu)  // global_addr[56:32]
          | (2u << 30);                           // type = 2 ("image")
    v8i g1;
    g1[0] = (int)(2u << 16);                      // data_size = 4 bytes
    g1[1] = (int)((nelem & 0xFFFFu) << 16);       // tensor_dim0[15:0]
    g1[2] = (int)((nelem >> 16) & 0xFFFFu);       // tensor_dim0[31:16]; tensor_dim1=0
    g1[3] = (int)((nelem & 0xFFFFu) << 16);       // tile_dim0 = nelem (<=16384 here)
    g1[4] = 0;                                    // tile_dim1/2 unused
    g1[5] = (int)nelem;                           // tensor_dim0_stride[31:0]
    g1[6] = 0;
    g1[7] = 0;
    v4i gz4  = {0, 0, 0, 0};
    v8i gz8  = {0, 0, 0, 0, 0, 0, 0, 0};
    // 6-arg toolchain form: (group0, group1, group2, group3, group4, cpol)
    __builtin_amdgcn_tensor_load_to_lds(g0, g1, gz4, gz4, gz8, 0);
}
#endif

// ---------------- fp32 WMMA GEMM ----------------
// Out[N, CT*16] = (RELU? relu(A) : A)[N,128] @ B[128, CT*16]
// Block = 256 threads = 8 waves; each wave owns one 16-row tile and keeps all
// CT column-tile accumulators in registers. W is staged once into LDS via TDM.
// A frag (16x4 f32): lanes 0-15 -> M=lane,{k,k+1}; lanes 16-31 -> M=lane-16,{k+2,k+3}
// B frag (4x16 f32): lanes 0-15 -> N=lane,{k,k+1}; lanes 16-31 -> N=lane-16,{k+2,k+3}
// C/D (16x16 f32): vgpr v -> M = v + 8*(lane>=16), N = lane&15

template <int CT, bool RELU>
__global__ void gcn_gemm_wmma(const float* __restrict__ A,
                              const float* __restrict__ B,
                              float* __restrict__ Out,
                              int n_rows) {
    constexpr int K  = 128;
    constexpr int NC = CT * 16;
    extern __shared__ float sW[];   // K * NC floats

#if __has_builtin(__builtin_amdgcn_tensor_load_to_lds)
    if (threadIdx.x < 32) {         // wave 0 issues one TDM copy of all of W
        tdm_load_linear_to_lds(B, (unsigned)(uintptr_t)sW, K * NC);
        __builtin_amdgcn_s_wait_tensorcnt(0);
    }
#else
    for (int i = threadIdx.x; i < K * NC; i += blockDim.x) sW[i] = B[i];
#endif
    __syncthreads();

    const int lane = threadIdx.x & 31;
    const int wv   = threadIdx.x >> 5;
    const int rt   = blockIdx.x * 8 + wv;       // this wave's 16-row tile
    const int row_tiles = n_rows >> 4;
    const int lr = lane & 15;
    const int kk = (lane >> 4) << 1;            // 0 or 2

    if (rt < row_tiles) {                       // wave-uniform guard: EXEC all-1s inside
        v8f acc[CT];
#pragma unroll
        for (int t = 0; t < CT; ++t) acc[t] = {};

        const float* arow = A + (size_t)(rt * 16 + lr) * K;
        for (int k = 0; k < K; k += 4) {
            v2f a = *(const v2f*)(arow + k + kk);     // global_load_b64
            if (RELU) { a.x = fmaxf(a.x, 0.0f); a.y = fmaxf(a.y, 0.0f); }
            const float* wrow0 = sW + (k + kk) * NC + lr;
#pragma unroll
            for (int t = 0; t < CT; ++t) {
                v2f b;
                b.x = wrow0[t * 16];
                b.y = wrow0[NC + t * 16];
                acc[t] = __builtin_amdgcn_wmma_f32_16x16x4_f32(
                    false, a, false, b, (short)0, acc[t], false, false);
            }
        }

        const int orow = rt * 16 + ((lane >> 4) << 3);
#pragma unroll
        for (int t = 0; t < CT; ++t) {
            float* op = Out + (size_t)orow * NC + t * 16 + lr;
#pragma unroll
            for (int v = 0; v < 8; ++v) op[(size_t)v * NC] = acc[t][v];
        }
    }
}

// ---------------- scatter phase ----------------

// out[i,c] = bias[c] + dinv[i]^2 * hw[i,c]   (self-loop term + bias)
__global__ void gcn_scatter_init(const float* __restrict__ hw,
                                 const float* __restrict__ dinv,
                                 const float* __restrict__ bias,
                                 float* __restrict__ out,
                                 int n, int ld_in, int c_out) {
    long long total  = (long long)n * c_out;
    long long i      = (long long)blockIdx.x * blockDim.x + threadIdx.x;
    long long stride = (long long)gridDim.x * blockDim.x;
    for (; i < total; i += stride) {
        int row = (int)(i / c_out);
        int col = (int)(i - (long long)row * c_out);
        float d = dinv[row];
        out[i] = bias[col] + d * d * hw[(size_t)row * ld_in + col];
    }
}

// one wave per edge: out[dst,:] += dinv[src]*dinv[dst] * hw[src,:]
__global__ void gcn_scatter_edges(const int* __restrict__ src,
                                  const int* __restrict__ dst,
                                  const float* __restrict__ dinv,
                                  const float* __restrict__ hw,
                                  float* out,
                                  int e, int ld_in, int c_out) {
    int w = (int)((blockIdx.x * (unsigned)blockDim.x + threadIdx.x) >> 5);
    int lane = threadIdx.x & 31;
    if (w >= e) return;
    int s = src[w], d = dst[w];
    float coef = dinv[s] * dinv[d];
    const float* hp = hw + (size_t)s * ld_in;
    float* op = out + (size_t)d * c_out;
    for (int c = lane; c < c_out; c += 32)
        atomicAdd(op + c, coef * hp[c]);
}

// ---------------- host launcher ----------------

extern "C" void kernel_launch(void* const* d_in, const int* in_sizes, int n_in,
                              void* d_out, int out_size, void* d_ws, size_t ws_size,
                              hipStream_t stream) {
    const float* x  = (const float*)d_in[0];
    const int*   ei = (const int*)d_in[1];
    const float* W1 = (const float*)d_in[2];
    const float* b1 = (const float*)d_in[3];
    const float* W2 = (const float*)d_in[4];
    const float* b2 = (const float*)d_in[5];
    const float* W3 = (const float*)d_in[6];
    const float* b3 = (const float*)d_in[7];

    const int D  = 128, H = 128;
    const int C  = in_sizes[7];          // 40
    const int N  = in_sizes[0] / D;      // 170000 (divisible by 16)
    const int E  = in_sizes[1] / 2;      // 1200000
    const int Cp = (C + 15) & ~15;       // 48

    char* ws = (char*)d_ws;
    size_t off = 0;
    float* dinv = (float*)(ws + off); off += ((size_t)N * 4 + 255) & ~(size_t)255;
    float* W3p  = (float*)(ws + off); off += ((size_t)H * Cp * 4 + 255) & ~(size_t)255;
    float* bufA = (float*)(ws + off); off += ((size_t)N * H * 4 + 255) & ~(size_t)255;
    float* bufB = (float*)(ws + off);

    const int* src = ei;        // edge_index[0]
    const int* dst = ei + E;    // edge_index[1]
    float* out = (float*)d_out;

    const int t = 256;
    gcn_init_deg <<<(N + t - 1) / t, t, 0, stream>>>(dinv, N);
    gcn_acc_deg  <<<(E + t - 1) / t, t, 0, stream>>>(dst, dinv, E);
    gcn_make_dinv<<<(N + t - 1) / t, t, 0, stream>>>(dinv, N);
    gcn_pad_w    <<<(H * Cp + t - 1) / t, t, 0, stream>>>(W3, W3p, H, C, Cp);

    const int row_tiles  = N >> 4;                    // 10625
    const int gemm_grid  = (row_tiles + 7) / 8;       // 8 waves/block
    const size_t sh128   = (size_t)128 * 128 * 4;     // 64 KB
    const size_t sh48    = (size_t)128 * 48 * 4;      // 24 KB

    const int sgrid = 4096;
    const int egrid = (int)(((long long)E * 32 + t - 1) / t);

    // layer 1: hw1 = x @ W1 ; out1 = segsum + b1
    gcn_gemm_wmma<8, false><<<gemm_grid, t, sh128, stream>>>(x, W1, bufA, N);
    gcn_scatter_init <<<sgrid, t, 0, stream>>>(bufA, dinv, b1, bufB, N, H, H);
    gcn_scatter_edges<<<egrid, t, 0, stream>>>(src, dst, dinv, bufA, bufB, E, H, H);

    // layer 2: hw2 = relu(out1) @ W2 (relu fused into A load)
    gcn_gemm_wmma<8, true><<<gemm_grid, t, sh128, stream>>>(bufB, W2, bufA, N);
    gcn_scatter_init <<<sgrid, t, 0, stream>>>(bufA, dinv, b2, bufB, N, H, H);
    gcn_scatter_edges<<<egrid, t, 0, stream>>>(src, dst, dinv, bufA, bufB, E, H, H);

    // layer 3: hw3 = relu(out2) @ W3pad (48 cols), scatter only 40 into d_out
    gcn_gemm_wmma<3, true><<<gemm_grid, t, sh48, stream>>>(bufB, W3p, bufA, N);
    gcn_scatter_init <<<sgrid, t, 0, stream>>>(bufA, dinv, b3, out, N, Cp, C);
    gcn_scatter_edges<<<egrid, t, 0, stream>>>(src, dst, dinv, bufA, out, E, Cp, C);
}